// ParallelEuclideanEmbed_76020921140203
// MI455X (gfx1250) — compile-verified
//
#include <hip/hip_runtime.h>
#include <math.h>

// ---------------- problem constants (from reference) ----------------
#define B_   16
#define T_   2048
#define I_   32
#define H_   256
#define E_   64
#define L_   3
#define DI_  512            // 2*H
#define N_   16
#define KC_  4              // conv taps
#define R_   16
#define NB_  4
#define MROWS (B_*T_)       // 32768

typedef float v2f __attribute__((ext_vector_type(2)));
typedef float v8f __attribute__((ext_vector_type(8)));
typedef int   v4i __attribute__((ext_vector_type(4)));

// ---------------- CDNA5 async global->LDS copy path ----------------
#if defined(__HIP_DEVICE_COMPILE__) && __has_builtin(__builtin_amdgcn_global_load_async_to_lds_b128)
#define HAVE_ASYNC_LDS 1
#else
#define HAVE_ASYNC_LDS 0
#endif

#define GLOBAL_AS __attribute__((address_space(1)))
#define LDS_AS    __attribute__((address_space(3)))

typedef GLOBAL_AS v4i* gv4i_p;
typedef LDS_AS    v4i* lv4i_p;

// copy 16 bytes global -> LDS. Async path: GLOBAL_LOAD_ASYNC_TO_LDS_B128
// (ASYNCcnt-tracked, no VGPR staging). Fallback: load+store.
__device__ __forceinline__ void cp_async16(const float* g, float* l)
{
#if HAVE_ASYNC_LDS
  __builtin_amdgcn_global_load_async_to_lds_b128((gv4i_p)g, (lv4i_p)l, 0, 0);
#else
  float4 v = *(const float4*)g;
  l[0] = v.x; l[1] = v.y; l[2] = v.z; l[3] = v.w;
#endif
}

__device__ __forceinline__ void wait_async0()
{
#if HAVE_ASYNC_LDS
#if __has_builtin(__builtin_amdgcn_s_wait_asynccnt)
  __builtin_amdgcn_s_wait_asynccnt(0);
#else
  asm volatile("s_wait_asynccnt 0x0" ::: "memory");
#endif
#endif
}

// =====================================================================
// gemm_big: C[M,N] = A[M,K] * W[N,K]^T (+bias[n]), f32 WMMA 16x16x4.
// Block tile 64(M) x 128(N), 8 waves arranged 2x4, wave tile 32x32
// (2x2 subtiles of 16x16). K staged through LDS in chunks of 32 with
// double-buffered async global->LDS copies overlapped with WMMA math.
// Requires M%64==0, N%128==0, K%32==0 (true for all uses here).
// =====================================================================
__global__ __launch_bounds__(256) void gemm_big(
    const float* __restrict__ A, const float* __restrict__ W,
    const float* __restrict__ bias, float* __restrict__ C,
    int M, int N, int Kd)
{
  const int BM = 64, BN = 128, KT = 32, LDK = KT + 4;  // stride 36: 16B-aligned rows, conflict-free
  __shared__ float As[2][BM * LDK];
  __shared__ float Bs[2][BN * LDK];

  const int tid  = threadIdx.x;
  const int lane = tid & 31;
  const int w    = tid >> 5;
  const int wm   = w & 1;         // 0..1  (M direction)
  const int wn   = w >> 1;        // 0..3  (N direction)
  const int m0   = blockIdx.x * BM;
  const int n0   = blockIdx.y * BN;
  const int half  = (lane < 16) ? 0 : 2;   // K offset per ISA f32 A/B layout
  const int idx16 = lane & 15;

  v8f acc[2][2] = {};

  const int NCH = Kd / KT;

  // issue one K-chunk's tiles (16B per async op)
  auto issue_tiles = [&](int chunk, int buf) {
    int k0 = chunk * KT;
    for (int i = tid; i < BM * (KT / 4); i += 256) {
      int r = i >> 3, c = (i & 7) << 2;
      cp_async16(A + (size_t)(m0 + r) * Kd + k0 + c, &As[buf][r * LDK + c]);
    }
    for (int i = tid; i < BN * (KT / 4); i += 256) {
      int r = i >> 3, c = (i & 7) << 2;
      cp_async16(W + (size_t)(n0 + r) * Kd + k0 + c, &Bs[buf][r * LDK + c]);
    }
  };

  issue_tiles(0, 0);

  for (int kt = 0; kt < NCH; ++kt) {
    const int cur = kt & 1;
    wait_async0();        // my async writes into buf[cur] done
    __syncthreads();      // everyone's done; prior readers of buf[cur^1] done too
    if (kt + 1 < NCH) issue_tiles(kt + 1, cur ^ 1);   // overlap with compute

    #pragma unroll
    for (int kk = 0; kk < KT; kk += 4) {
      v2f a[2], bf[2];
      #pragma unroll
      for (int mi = 0; mi < 2; ++mi) {
        const float* p = &As[cur][(wm * 32 + mi * 16 + idx16) * LDK + kk + half];
        a[mi].x = p[0]; a[mi].y = p[1];
      }
      #pragma unroll
      for (int ni = 0; ni < 2; ++ni) {
        const float* p = &Bs[cur][(wn * 32 + ni * 16 + idx16) * LDK + kk + half];
        bf[ni].x = p[0]; bf[ni].y = p[1];
      }
      #pragma unroll
      for (int mi = 0; mi < 2; ++mi)
        #pragma unroll
        for (int ni = 0; ni < 2; ++ni)
          acc[mi][ni] = __builtin_amdgcn_wmma_f32_16x16x4_f32(
              false, a[mi], false, bf[ni], (short)0, acc[mi][ni], false, false);
    }
  }

  // epilogue per ISA 16x16 f32 C/D layout
  #pragma unroll
  for (int mi = 0; mi < 2; ++mi) {
    #pragma unroll
    for (int ni = 0; ni < 2; ++ni) {
      int col = n0 + wn * 32 + ni * 16 + idx16;
      float bb = bias ? bias[col] : 0.0f;
      #pragma unroll
      for (int j = 0; j < 8; ++j) {
        int row = m0 + wm * 32 + mi * 16 + ((lane < 16) ? j : j + 8);
        C[(size_t)row * N + col] = acc[mi][ni][j] + bb;
      }
    }
  }
}

// =====================================================================
// gemm_small48: C[M,48] = A[M,K] * W[48,K]^T  (the dbc projection).
// Block tile 128(M) x 48(N): 8 waves, one 16-row strip per wave,
// 3 accumulators covering the 48 columns. Same async double buffering.
// =====================================================================
__global__ __launch_bounds__(256) void gemm_small48(
    const float* __restrict__ A, const float* __restrict__ W,
    float* __restrict__ C, int Kd)
{
  const int BM = 128, BN = 48, KT = 32, LDK = KT + 4;
  __shared__ float As[2][BM * LDK];
  __shared__ float Bs[2][BN * LDK];

  const int tid  = threadIdx.x;
  const int lane = tid & 31;
  const int w    = tid >> 5;
  const int m0   = blockIdx.x * BM;
  const int half  = (lane < 16) ? 0 : 2;
  const int idx16 = lane & 15;

  v8f acc[3] = {};

  const int NCH = Kd / KT;

  auto issue_tiles = [&](int chunk, int buf) {
    int k0 = chunk * KT;
    for (int i = tid; i < BM * (KT / 4); i += 256) {
      int r = i >> 3, c = (i & 7) << 2;
      cp_async16(A + (size_t)(m0 + r) * Kd + k0 + c, &As[buf][r * LDK + c]);
    }
    for (int i = tid; i < BN * (KT / 4); i += 256) {
      int r = i >> 3, c = (i & 7) << 2;
      cp_async16(W + (size_t)r * Kd + k0 + c, &Bs[buf][r * LDK + c]);
    }
  };

  issue_tiles(0, 0);

  for (int kt = 0; kt < NCH; ++kt) {
    const int cur = kt & 1;
    wait_async0();
    __syncthreads();
    if (kt + 1 < NCH) issue_tiles(kt + 1, cur ^ 1);

    #pragma unroll
    for (int kk = 0; kk < KT; kk += 4) {
      v2f a;
      const float* pa = &As[cur][(w * 16 + idx16) * LDK + kk + half];
      a.x = pa[0]; a.y = pa[1];
      #pragma unroll
      for (int ni = 0; ni < 3; ++ni) {
        v2f bf;
        const float* pb = &Bs[cur][(ni * 16 + idx16) * LDK + kk + half];
        bf.x = pb[0]; bf.y = pb[1];
        acc[ni] = __builtin_amdgcn_wmma_f32_16x16x4_f32(
            false, a, false, bf, (short)0, acc[ni], false, false);
      }
    }
  }

  #pragma unroll
  for (int ni = 0; ni < 3; ++ni) {
    int col = ni * 16 + idx16;
    #pragma unroll
    for (int j = 0; j < 8; ++j) {
      int row = m0 + w * 16 + ((lane < 16) ? j : j + 8);
      C[(size_t)row * 48 + col] = acc[ni][j];
    }
  }
}

// =====================================================================
// Depthwise causal conv (K=4) + bias + SiLU on the x-half of xz.
// One thread per (b,t,d). DI*T = 2^20 -> cheap index math.
// =====================================================================
__global__ __launch_bounds__(256) void conv_silu(
    const float* __restrict__ xz, const float* __restrict__ cw,
    const float* __restrict__ cb, float* __restrict__ xc)
{
  int idx = blockIdx.x * 256 + threadIdx.x;
  int d = idx & (DI_ - 1);
  int t = (idx >> 9) & (T_ - 1);
  int b = idx >> 20;
  float acc = cb[d];
  const float* wk = cw + d * KC_;
  #pragma unroll
  for (int k = 0; k < KC_; ++k) {
    int tt = t - (KC_ - 1) + k;
    if (tt >= 0)
      acc += wk[k] * xz[((size_t)(b * T_ + tt)) * (2 * DI_) + d];
  }
  xc[idx] = acc / (1.0f + __expf(-acc));      // silu
}

// =====================================================================
// Selective scan. Thread layout: lane-group of 16 = one (b,d) channel,
// lane-within-group = state index n. 131072 threads = 4096 waves.
// delta is recomputed inline (lane r supplies dt[r]*dt_w[d,r] partial),
// both reductions are 16-lane xor-butterflies (wave32 safe, masks<=8).
// =====================================================================
__global__ __launch_bounds__(256) void scan_kernel(
    const float* __restrict__ dbc, const float* __restrict__ xc,
    const float* __restrict__ xz,  const float* __restrict__ dtw,
    const float* __restrict__ dtb, const float* __restrict__ alog,
    const float* __restrict__ Dp,  float* __restrict__ yg)
{
  int tid = threadIdx.x;
  int n = tid & 15;
  int c = blockIdx.x * 16 + (tid >> 4);
  int b = c >> 9;                 // / DI_
  int d = c & (DI_ - 1);

  float An    = -__expf(alog[d * N_ + n]);
  float wtr   = dtw[d * R_ + n];  // this lane's r == n partial weight
  float biasd = dtb[d];
  float Dd    = Dp[d];
  float hstate = 0.0f;

  for (int t = 0; t < T_; ++t) {
    size_t row = (size_t)(b * T_ + t);
    const float* drow = dbc + row * 48;

    // delta = softplus( dt . dt_w[d] + dt_b[d] )
    float part = drow[n] * wtr;
    part += __shfl_xor(part, 1);
    part += __shfl_xor(part, 2);
    part += __shfl_xor(part, 4);
    part += __shfl_xor(part, 8);
    float x = part + biasd;
    float delta = (x > 20.0f) ? x : log1pf(__expf(x));

    float u  = xc[row * DI_ + d];
    float Bn = drow[R_ + n];
    float Cn = drow[R_ + N_ + n];

    hstate = __expf(delta * An) * hstate + (delta * u) * Bn;

    float yp = hstate * Cn;
    yp += __shfl_xor(yp, 1);
    yp += __shfl_xor(yp, 2);
    yp += __shfl_xor(yp, 4);
    yp += __shfl_xor(yp, 8);

    if (n == 0) {
      float y = yp + u * Dd;
      float z = xz[row * (2 * DI_) + DI_ + d];
      yg[row * DI_ + d] = y * (z / (1.0f + __expf(-z)));
    }
  }
}

// mean over T: m[b,h] = (1/T) sum_t h[b,t,h]
__global__ __launch_bounds__(256) void mean_kernel(
    const float* __restrict__ hbuf, float* __restrict__ m)
{
  int idx = blockIdx.x * 256 + threadIdx.x;   // b*H + h
  int b  = idx >> 8;
  int hh = idx & 255;
  float acc = 0.0f;
  for (int t = 0; t < T_; ++t)
    acc += hbuf[((size_t)(b * T_ + t)) * H_ + hh];
  m[idx] = acc * (1.0f / T_);
}

// head: e[nb,b,e] = m[nb,b,:] . op_w[nb,e,:] + op_b[nb,e]; out = 4 branches + sum
__global__ __launch_bounds__(256) void head_kernel(
    const float* __restrict__ m, const float* __restrict__ opw,
    const float* __restrict__ opb, float* __restrict__ out)
{
  int idx = blockIdx.x * 256 + threadIdx.x;   // b*E + e
  int b = idx / E_;
  int e = idx % E_;
  float s = 0.0f;
  for (int nb = 0; nb < NB_; ++nb) {
    float acc = opb[nb * E_ + e];
    const float* mv = m + (size_t)nb * B_ * H_ + (size_t)b * H_;
    const float* wv = opw + (size_t)nb * E_ * H_ + (size_t)e * H_;
    for (int hh = 0; hh < H_; ++hh) acc += mv[hh] * wv[hh];
    out[(size_t)nb * B_ * E_ + idx] = acc;
    s += acc;
  }
  out[(size_t)4 * B_ * E_ + idx] = s;
}

// =====================================================================
extern "C" void kernel_launch(void* const* d_in, const int* in_sizes, int n_in,
                              void* d_out, int out_size, void* d_ws, size_t ws_size,
                              hipStream_t stream)
{
  (void)in_sizes; (void)n_in; (void)out_size; (void)ws_size;

  const float* xin[4] = {(const float*)d_in[0], (const float*)d_in[1],
                         (const float*)d_in[2], (const float*)d_in[3]};
  const float* ip_w  = (const float*)d_in[4];
  const float* ip_b  = (const float*)d_in[5];
  const float* in_w  = (const float*)d_in[6];
  const float* cv_w  = (const float*)d_in[7];
  const float* cv_b  = (const float*)d_in[8];
  const float* xp_w  = (const float*)d_in[9];
  const float* dt_w  = (const float*)d_in[10];
  const float* dt_b  = (const float*)d_in[11];
  const float* A_log = (const float*)d_in[12];
  const float* Dp    = (const float*)d_in[13];
  const float* om_w  = (const float*)d_in[14];
  const float* op_w  = (const float*)d_in[15];
  const float* op_b  = (const float*)d_in[16];
  float* out = (float*)d_out;

  float* ws = (float*)d_ws;
  size_t off = 0;
  float* h_a  = ws + off; off += (size_t)MROWS * H_;
  float* h_b  = ws + off; off += (size_t)MROWS * H_;
  float* xz   = ws + off; off += (size_t)MROWS * (2 * DI_);
  float* xc   = ws + off; off += (size_t)MROWS * DI_;
  float* dbc  = ws + off; off += (size_t)MROWS * 48;
  float* yg   = ws + off; off += (size_t)MROWS * DI_;
  float* mbuf = ws + off; off += (size_t)NB_ * B_ * H_;

  for (int nb = 0; nb < NB_; ++nb) {
    // input projection: h = x @ ip_w^T + ip_b   (M=32768, N=256, K=32)
    gemm_big<<<dim3(MROWS / 64, H_ / 128), 256, 0, stream>>>(
        xin[nb], ip_w + (size_t)nb * H_ * I_, ip_b + (size_t)nb * H_,
        h_a, MROWS, H_, I_);

    float* cur = h_a;
    float* nxt = h_b;
    for (int l = 0; l < L_; ++l) {
      size_t bl = (size_t)nb * L_ + l;

      // xz = h @ in_w^T   (N=1024, K=256)
      gemm_big<<<dim3(MROWS / 64, (2 * DI_) / 128), 256, 0, stream>>>(
          cur, in_w + bl * (2 * DI_) * H_, nullptr, xz, MROWS, 2 * DI_, H_);

      // depthwise causal conv + silu
      conv_silu<<<(MROWS * DI_) / 256, 256, 0, stream>>>(
          xz, cv_w + bl * DI_ * KC_, cv_b + bl * DI_, xc);

      // dbc = xc @ xp_w^T   (N=48, K=512)
      gemm_small48<<<MROWS / 128, 256, 0, stream>>>(
          xc, xp_w + bl * 48 * DI_, dbc, DI_);

      // selective scan + skip + output gating
      scan_kernel<<<(B_ * DI_) / 16, 256, 0, stream>>>(
          dbc, xc, xz,
          dt_w + bl * DI_ * R_, dt_b + bl * DI_,
          A_log + bl * DI_ * N_, Dp + bl * DI_, yg);

      // h_next = yg @ om_w^T   (N=256, K=512)
      gemm_big<<<dim3(MROWS / 64, H_ / 128), 256, 0, stream>>>(
          yg, om_w + bl * H_ * DI_, nullptr, nxt, MROWS, H_, DI_);

      float* tmp = cur; cur = nxt; nxt = tmp;
    }

    mean_kernel<<<(B_ * H_) / 256, 256, 0, stream>>>(
        cur, mbuf + (size_t)nb * B_ * H_);
  }

  head_kernel<<<(B_ * E_) / 256, 256, 0, stream>>>(mbuf, op_w, op_b, out);
}